// FTScanWithAdditionalInputs_56298431316005
// MI455X (gfx1250) — compile-verified
//
#include <hip/hip_runtime.h>

typedef __attribute__((ext_vector_type(2))) float v2f;
typedef __attribute__((ext_vector_type(8))) float v8f;

#define T_TOTAL 8192
#define NCOL    8192            // 4 * 2048 flattened columns
#define CHUNK   256             // timesteps per chunk
#define NCHUNK  (T_TOTAL / CHUNK)   // 32
#define TILE    16
#define NTILE   (CHUNK / TILE)      // 16

// ---------------------------------------------------------------------------
// Kernel 1: per-(chunk, column) scaled partial sums.
// One thread per (chunk, col): 32 * 8192 = 262144 threads. Loads are fully
// coalesced across the wave (consecutive cols). Prefetch hints ahead.
// ---------------------------------------------------------------------------
__global__ void k_chunk_sums(const float* __restrict__ xs,
                             const float* __restrict__ scale,
                             float* __restrict__ sums) {
  const int tid   = blockIdx.x * blockDim.x + threadIdx.x;
  const int col   = tid & (NCOL - 1);
  const int chunk = tid >> 13;            // NCOL == 2^13
  const float s = scale[0];
  const float* p = xs + (size_t)chunk * CHUNK * NCOL + col;
  float acc = 0.0f;
  for (int tt = 0; tt < CHUNK; tt += 16) {
    __builtin_prefetch(p + (size_t)48 * NCOL, 0, 1);  // speculative; OOB dropped
#pragma unroll
    for (int t = 0; t < 16; ++t) acc += p[(size_t)t * NCOL];
    p += (size_t)16 * NCOL;
  }
  sums[(size_t)chunk * NCOL + col] = acc * s;
}

// ---------------------------------------------------------------------------
// Kernel 2: in-place exclusive scan of the 32 chunk sums per column; the
// final running value is the full column sum == carry output.
// ---------------------------------------------------------------------------
__global__ void k_scan_offsets(float* __restrict__ sums,
                               float* __restrict__ carry) {
  const int col = blockIdx.x * blockDim.x + threadIdx.x;
  float run = 0.0f;
#pragma unroll
  for (int c = 0; c < NCHUNK; ++c) {
    const float v = sums[(size_t)c * NCOL + col];
    sums[(size_t)c * NCOL + col] = run;   // exclusive prefix (chunk offset)
    run += v;
  }
  carry[col] = run;
}

// ---------------------------------------------------------------------------
// Kernel 3: WMMA tile scan. One wave handles one (chunk, 16-column tile):
// 512 col-tiles * 32 chunks = 16384 waves. For each 16x16 (t x col) tile X:
//     S = (scale * L) x X    via 4 chained V_WMMA_F32_16X16X4_F32 (K=4)
// where L is the lower-triangular ones matrix -> S is the exact fp32
// inclusive scan of the tile, pre-scaled. Add the running column offset,
// store, and advance the offset by tile row 15 (broadcast via shfl).
// ---------------------------------------------------------------------------
__global__ void k_scan_wmma(const float* __restrict__ xs,
                            const float* __restrict__ scale,
                            const float* __restrict__ offs,
                            float* __restrict__ ys) {
  const int lane  = threadIdx.x & 31;
  const int wave  = threadIdx.x >> 5;
  const int w     = blockIdx.x * (blockDim.x >> 5) + wave;
  const int ctile = w & 511;              // 512 column tiles
  const int chunk = w >> 9;               // 32 chunks
  const int q     = lane & 15;
  const int half  = lane >> 4;
  const int col   = ctile * 16 + q;
  const float s   = scale[0];

  // A_k = scale * L[:, 4k + 2*half + v], row index q  (constant per lane)
  v2f a[4];
#pragma unroll
  for (int k = 0; k < 4; ++k) {
    const int j0 = 4 * k + 2 * half;
    a[k].x = (j0     <= q) ? s : 0.0f;
    a[k].y = (j0 + 1 <= q) ? s : 0.0f;
  }

  float off = offs[(size_t)chunk * NCOL + col];

  const float* xcol = xs + col;
  float*       ycol = ys + col;
  int t_base = chunk * CHUNK;

  for (int tile = 0; tile < NTILE; ++tile) {
    const float* xt = xcol + (size_t)t_base * NCOL;
    // B_k: rows t_local = 4k + 2*half + v of the tile, column q
    v2f b[4];
#pragma unroll
    for (int k = 0; k < 4; ++k) {
      const int t0 = 4 * k + 2 * half;
      b[k].x = xt[(size_t)t0 * NCOL];
      b[k].y = xt[(size_t)(t0 + 1) * NCOL];
    }

    v8f c = {};
#pragma unroll
    for (int k = 0; k < 4; ++k) {
      // 8 args: (neg_a, A, neg_b, B, c_mod, C, reuse_a, reuse_b)
      c = __builtin_amdgcn_wmma_f32_16x16x4_f32(false, a[k], false, b[k],
                                                (short)0, c, false, false);
    }

    // S[15][q] (scaled tile column sum) lives in lane 16+q, VGPR 7.
    const float colsum = __shfl(c[7], 16 + q, 32);

    float* yt = ycol + (size_t)t_base * NCOL;
#pragma unroll
    for (int v = 0; v < 8; ++v) {
      yt[(size_t)(v + 8 * half) * NCOL] = c[v] + off;   // row m = v + 8*half
    }
    off += colsum;
    t_base += TILE;
  }
}

// ---------------------------------------------------------------------------
// d_in[0] = xs (8192*4*2048 f32), d_in[1] = scale (1 f32)
// d_out   = [carry (8192 f32) | ys (8192*4*2048 f32)]
// d_ws    = chunk sums / offsets (NCHUNK * NCOL f32 = 1 MiB)
// ---------------------------------------------------------------------------
extern "C" void kernel_launch(void* const* d_in, const int* in_sizes, int n_in,
                              void* d_out, int out_size, void* d_ws, size_t ws_size,
                              hipStream_t stream) {
  (void)in_sizes; (void)n_in; (void)out_size; (void)ws_size;
  const float* xs    = (const float*)d_in[0];
  const float* scale = (const float*)d_in[1];
  float* carry = (float*)d_out;
  float* ys    = (float*)d_out + NCOL;
  float* sums  = (float*)d_ws;

  k_chunk_sums  <<<(NCHUNK * NCOL) / 256, 256, 0, stream>>>(xs, scale, sums);
  k_scan_offsets<<<NCOL / 256,            256, 0, stream>>>(sums, carry);
  k_scan_wmma   <<<(512 * NCHUNK) / 8,    256, 0, stream>>>(xs, scale, sums, ys);
}